// TagWingLoss_87136296501795
// MI455X (gfx1250) — compile-verified
//
#include <hip/hip_runtime.h>
#include <hip/hip_bf16.h>
#include <math.h>

// ---------------------------------------------------------------------------
// TagWingLoss for MI455X (gfx1250, wave32).
// Two bandwidth-bound passes over pred/gt (≈320MB => ~14us at 23.3 TB/s),
// then a per-image finalize that computes the 128x128 Gram matrix of the
// group-mean tags with V_WMMA_F32_16X16X4_F32 (T is 128x4 f32: a perfect
// 16x16x4 f32 WMMA tiling) for the push term.
// ---------------------------------------------------------------------------

#define IMG     4
#define NANCH   2000000
#define NSEG    129          // NUM_GROUPS + 1
#define NBLK    128          // partial-reduction blocks per image (both passes)
#define CHUNK   (NANCH/NBLK) // 15625 anchors per block
#define TPB     256

#define ALPHA_F 10.0f
#define INV_BETA 0.5f
// C = ALPHA - ALPHA*log(1 + ALPHA/BETA) = 10 - 10*ln(6)
#define C_WING  (-7.9175946922805550f)
#define EPS_F   1e-6f

typedef float v2f __attribute__((ext_vector_type(2)));
typedef float v8f __attribute__((ext_vector_type(8)));

// ---------------- Pass 1: segmented counts + sums (partials per block) ------
__global__ void k_pass1(const float4* __restrict__ pred,
                        const int*    __restrict__ gt,
                        unsigned*     __restrict__ p_cnt,   // [IMG][NBLK][NSEG]
                        float*        __restrict__ p_sum) { // [IMG][NBLK][NSEG][4]
    __shared__ unsigned s_cnt[NSEG];
    __shared__ float    s_sum[NSEG * 4];
    const int tid = threadIdx.x;
    const int blk = blockIdx.x;
    const int img = blockIdx.y;

    for (int i = tid; i < NSEG; i += TPB) s_cnt[i] = 0u;
    for (int i = tid; i < NSEG * 4; i += TPB) s_sum[i] = 0.0f;
    __syncthreads();

    const float4* pp = pred + (size_t)img * NANCH;
    const int*    gg = gt   + (size_t)img * NANCH;
    const int start = blk * CHUNK;
    const int end   = start + CHUNK;
    for (int i = start + tid; i < end; i += TPB) {
        int seg = gg[i] + 1;
        seg = (seg < 0) ? 0 : (seg > 128 ? 128 : seg);
        float4 v = pp[i];
        atomicAdd(&s_cnt[seg], 1u);
        atomicAdd(&s_sum[seg * 4 + 0], v.x);
        atomicAdd(&s_sum[seg * 4 + 1], v.y);
        atomicAdd(&s_sum[seg * 4 + 2], v.z);
        atomicAdd(&s_sum[seg * 4 + 3], v.w);
    }
    __syncthreads();

    const size_t base = ((size_t)img * NBLK + blk) * NSEG;
    for (int s = tid; s < NSEG; s += TPB)      p_cnt[base + s] = s_cnt[s];
    for (int i = tid; i < NSEG * 4; i += TPB)  p_sum[base * 4 + i] = s_sum[i];
}

// ---------------- Reduce 1: counts + means (fixed order => deterministic) ---
__global__ void k_reduce1(const unsigned* __restrict__ p_cnt,
                          const float*    __restrict__ p_sum,
                          float* __restrict__ counts,   // [IMG][NSEG]
                          float* __restrict__ means) {  // [IMG][NSEG][4]
    const int img = blockIdx.x;
    const int s   = threadIdx.x;
    if (s >= NSEG) return;
    unsigned c = 0u;
    float s0 = 0.f, s1 = 0.f, s2 = 0.f, s3 = 0.f;
    for (int b = 0; b < NBLK; ++b) {
        const size_t base = ((size_t)img * NBLK + b) * NSEG;
        c += p_cnt[base + s];
        const float* ps = p_sum + (base + s) * 4;
        s0 += ps[0]; s1 += ps[1]; s2 += ps[2]; s3 += ps[3];
    }
    const float cf  = (float)c;
    const float den = fmaxf(cf, 1.0f);
    counts[img * NSEG + s] = cf;
    float* m = means + ((size_t)img * NSEG + s) * 4;
    m[0] = s0 / den; m[1] = s1 / den; m[2] = s2 / den; m[3] = s3 / den;
}

// ---------------- Pass 2: segmented wing-loss sums --------------------------
__global__ void k_pass2(const float4* __restrict__ pred,
                        const int*    __restrict__ gt,
                        const float*  __restrict__ means,
                        float*        __restrict__ p_wl) { // [IMG][NBLK][NSEG]
    __shared__ float s_mean[NSEG * 4];
    __shared__ float s_wl[NSEG];
    const int tid = threadIdx.x;
    const int blk = blockIdx.x;
    const int img = blockIdx.y;

    for (int i = tid; i < NSEG * 4; i += TPB)
        s_mean[i] = means[(size_t)img * NSEG * 4 + i];
    for (int i = tid; i < NSEG; i += TPB) s_wl[i] = 0.0f;
    __syncthreads();

    const float4* pp = pred + (size_t)img * NANCH;
    const int*    gg = gt   + (size_t)img * NANCH;
    const int start = blk * CHUNK;
    const int end   = start + CHUNK;
    for (int i = start + tid; i < end; i += TPB) {
        int seg = gg[i] + 1;
        seg = (seg < 0) ? 0 : (seg > 128 ? 128 : seg);
        float4 v = pp[i];
        const float* m = &s_mean[seg * 4];
        float w = 0.0f;
        float d;
        d = fabsf(v.x - m[0]); w += (d < ALPHA_F) ? ALPHA_F * log1pf(d * INV_BETA) : d - C_WING;
        d = fabsf(v.y - m[1]); w += (d < ALPHA_F) ? ALPHA_F * log1pf(d * INV_BETA) : d - C_WING;
        d = fabsf(v.z - m[2]); w += (d < ALPHA_F) ? ALPHA_F * log1pf(d * INV_BETA) : d - C_WING;
        d = fabsf(v.w - m[3]); w += (d < ALPHA_F) ? ALPHA_F * log1pf(d * INV_BETA) : d - C_WING;
        atomicAdd(&s_wl[seg], w);
    }
    __syncthreads();

    const size_t base = ((size_t)img * NBLK + blk) * NSEG;
    for (int s = tid; s < NSEG; s += TPB) p_wl[base + s] = s_wl[s];
}

// ---------------- Finalize per image: pull + WMMA push ----------------------
__global__ void k_finalize(const float* __restrict__ p_wl,
                           const float* __restrict__ counts,
                           const float* __restrict__ means,
                           float*       __restrict__ losses) { // [IMG]
    __shared__ float    s_wl[NSEG];
    __shared__ float    s_tags[128 * 4];
    __shared__ float    s_q[128];
    __shared__ float    s_valid[128];
    __shared__ float    s_red[TPB];
    __shared__ unsigned s_num;
    __shared__ float    s_pull;

    const int tid = threadIdx.x;
    const int img = blockIdx.x;
    if (tid == 0) s_num = 0u;

    // deterministic fixed-order reduction of wl partials
    if (tid < NSEG) {
        float acc = 0.0f;
        for (int b = 0; b < NBLK; ++b)
            acc += p_wl[((size_t)img * NBLK + b) * NSEG + tid];
        s_wl[tid] = acc;
    }
    __syncthreads();

    float pullv = 0.0f;
    if (tid < 128) {
        const int s = tid + 1;
        const float c = counts[img * NSEG + s];
        const float valid = (c > 0.0f) ? 1.0f : 0.0f;
        const float* m = means + ((size_t)img * NSEG + s) * 4;
        const float t0 = m[0], t1 = m[1], t2 = m[2], t3 = m[3];
        s_tags[tid * 4 + 0] = t0; s_tags[tid * 4 + 1] = t1;
        s_tags[tid * 4 + 2] = t2; s_tags[tid * 4 + 3] = t3;
        s_q[tid]     = t0 * t0 + t1 * t1 + t2 * t2 + t3 * t3;
        s_valid[tid] = valid;
        if (valid > 0.0f) {
            atomicAdd(&s_num, 1u);
            pullv = s_wl[s] / fmaxf(c * 4.0f, 1.0f);
        }
    }
    s_red[tid] = pullv;
    __syncthreads();
    for (int off = TPB / 2; off > 0; off >>= 1) {
        if (tid < off) s_red[tid] += s_red[tid + off];
        __syncthreads();
    }
    if (tid == 0) s_pull = s_red[0];
    __syncthreads();

    // ---- push term: G = T * T^T via v_wmma_f32_16x16x4_f32 -----------------
    // 8 waves; wave ti owns tile-row ti; loop tiles tj=0..7. Uniform control
    // flow => EXEC all ones around every WMMA.
    const int lane = tid & 31;
    const int ti   = tid >> 5;       // wave id = tile row
    const int ln   = lane & 15;
    const int kh   = lane >> 4;      // K-pair half select (0 or 1)

    v2f a;
    a.x = s_tags[(ti * 16 + ln) * 4 + kh * 2 + 0];
    a.y = s_tags[(ti * 16 + ln) * 4 + kh * 2 + 1];

    float psum = 0.0f;
    for (int tj = 0; tj < 8; ++tj) {
        v2f b;
        b.x = s_tags[(tj * 16 + ln) * 4 + kh * 2 + 0];
        b.y = s_tags[(tj * 16 + ln) * 4 + kh * 2 + 1];
        v8f acc = {0.f, 0.f, 0.f, 0.f, 0.f, 0.f, 0.f, 0.f};
        acc = __builtin_amdgcn_wmma_f32_16x16x4_f32(
            false, a, false, b, (short)0, acc, false, false);
        #pragma unroll
        for (int v = 0; v < 8; ++v) {
            const int i = ti * 16 + v + 8 * kh;   // D layout: M = v + 8*(lane>=16)
            const int j = tj * 16 + ln;           //           N = lane & 15
            const float d2 = s_q[i] + s_q[j] - 2.0f * acc[v];
            psum += __expf(-d2) * s_valid[i] * s_valid[j];
        }
    }
    __syncthreads();
    s_red[tid] = psum;
    __syncthreads();
    for (int off = TPB / 2; off > 0; off >>= 1) {
        if (tid < off) s_red[tid] += s_red[tid + off];
        __syncthreads();
    }

    if (tid == 0) {
        const float num  = (float)s_num;
        const float Ptot = s_red[0];
        const float push = (Ptot - num) / ((num - 1.0f) * num + EPS_F) * 0.5f;
        const float pull = s_pull / (num + EPS_F);
        losses[img] = push + pull;
    }
}

// ---------------- Final mean over images ------------------------------------
__global__ void k_final(const float* __restrict__ losses, float* __restrict__ out) {
    if (threadIdx.x == 0 && blockIdx.x == 0)
        out[0] = 0.25f * (losses[0] + losses[1] + losses[2] + losses[3]);
}

// ---------------------------------------------------------------------------
extern "C" void kernel_launch(void* const* d_in, const int* in_sizes, int n_in,
                              void* d_out, int out_size, void* d_ws, size_t ws_size,
                              hipStream_t stream) {
    const float4* pred = (const float4*)d_in[0];
    const int*    gt   = (const int*)d_in[1];
    // d_in[2] (anchor_inds) is unused by the reference loss.

    // Workspace layout (32-bit element offsets), total ~1.6 MB:
    float*    wsf    = (float*)d_ws;
    unsigned* p_cnt  = (unsigned*)d_ws;            // 4*128*129         = 66048
    float*    p_sum  = wsf + 66048;                // 4*128*129*4       = 264192
    float*    p_wl   = wsf + 330240;               // 4*128*129         = 66048
    float*    counts = wsf + 396288;               // 4*129             = 516
    float*    means  = wsf + 396804;               // 4*129*4           = 2064
    float*    losses = wsf + 398868;               // 4

    dim3 grid1(NBLK, IMG);
    k_pass1   <<<grid1, TPB, 0, stream>>>(pred, gt, p_cnt, p_sum);
    k_reduce1 <<<IMG,   TPB, 0, stream>>>(p_cnt, p_sum, counts, means);
    k_pass2   <<<grid1, TPB, 0, stream>>>(pred, gt, means, p_wl);
    k_finalize<<<IMG,   TPB, 0, stream>>>(p_wl, counts, means, losses);
    k_final   <<<1,      32, 0, stream>>>(losses, (float*)d_out);
}